// SparseDecoder_41781441855727
// MI455X (gfx1250) — compile-verified
//
#include <hip/hip_runtime.h>

// SparseDecoder for MI455X (gfx1250, wave32).
// fp32 WMMA 16x16x4 for the per-kernel-offset GEMMs; L2-resident gathers;
// non-returning global_atomic_add_f32 scatters; LDS-atomic BN reduction.

typedef float v2f __attribute__((ext_vector_type(2)));
typedef float v8f __attribute__((ext_vector_type(8)));

#define KSTEN 9
#define BN_EPS 1e-5f

// One wave handles one (k, 16-row) tile: computes a 16 x COUT contribution
// block via V_WMMA_F32_16X16X4_F32 and scatter-adds it with f32 atomics.
// A layout (16x4 f32): lanes 0-15 -> {K=0,K=1}, lanes 16-31 -> {K=2,K=3}.
// B layout (4x16 f32): mirrored (lanes 0-15 rows K=0/1, lanes 16-31 rows K=2/3).
template<int CIN, int COUT>
__global__ __launch_bounds__(256) void up_gemm_scatter(
    const float* __restrict__ x,       // [*, CIN] gather source
    const float* __restrict__ W,       // [KSTEN, CIN, COUT]
    const int*   __restrict__ in_idx,  // [KSTEN, P]
    const int*   __restrict__ out_idx, // [KSTEN, P]
    float*       __restrict__ out,     // [Nout, COUT] accumulator (pre-zeroed)
    int P)                             // multiple of 16
{
  constexpr int NT = COUT / 16;
  const int lane = threadIdx.x & 31;
  const int half = lane >> 4;   // 0: K=0,1 | 1: K=2,3
  const int l16  = lane & 15;
  const int wavesPerBlock = blockDim.x >> 5;
  const int tile = blockIdx.x * wavesPerBlock + (threadIdx.x >> 5);
  const int tilesPerK = P >> 4;
  const int k = tile / tilesPerK;
  if (k >= KSTEN) return;                       // wave-uniform guard: EXEC stays all-ones
  const int p0 = (tile - k * tilesPerK) << 4;

  const int* iidx_k = in_idx  + (size_t)k * P;
  const int* oidx_k = out_idx + (size_t)k * P;
  const int row  = iidx_k[p0 + l16];            // gather row for M = l16
  const int orow = oidx_k[p0 + l16];            // scatter row for M = l16

  const float* __restrict__ Wk = W + (size_t)k * CIN * COUT;
  const float* __restrict__ xr = x + (size_t)row * CIN + half * 2;

  v8f acc[NT] = {};
  #pragma unroll 4
  for (int c0 = 0; c0 < CIN; c0 += 4) {
    v2f a = *(const v2f*)(xr + c0);             // 8B-aligned global_load_b64
    const float* wb = Wk + (size_t)(c0 + half * 2) * COUT + l16;
    #pragma unroll
    for (int nt = 0; nt < NT; ++nt) {
      v2f b;
      b.x = wb[nt * 16];
      b.y = wb[nt * 16 + COUT];
      acc[nt] = __builtin_amdgcn_wmma_f32_16x16x4_f32(
          false, a, false, b, (short)0, acc[nt], false, false);
    }
  }

  // D tile: lane holds D[M = r + 8*half][N = l16] in acc[nt][r].
  #pragma unroll
  for (int r = 0; r < 8; ++r) {
    const int m = r + half * 8;
    const int drow = __shfl(orow, m, 32);       // ds_bpermute broadcast of out row
    float* op = out + (size_t)drow * COUT + l16;
    #pragma unroll
    for (int nt = 0; nt < NT; ++nt)
      atomicAdd(op + nt * 16, acc[nt][r]);      // non-returning global_atomic_add_f32
  }
}

// Per-channel sum / sum-of-squares over all N rows (grid-stride),
// LDS float-atomic block reduction, then one global atomic per channel.
template<int C>
__global__ __launch_bounds__(256) void bn_stats(
    const float* __restrict__ buf, int N, float* __restrict__ stats /* [2C] */)
{
  __shared__ float s_acc[2 * C];
  if (threadIdx.x < 2 * C) s_acc[threadIdx.x] = 0.f;
  __syncthreads();
  const int c   = threadIdx.x & (C - 1);
  const int rpb = 256 / C;
  float sum = 0.f, sq = 0.f;
  for (long row = (long)blockIdx.x * rpb + (threadIdx.x / C); row < N;
       row += (long)gridDim.x * rpb) {
    float v = buf[row * C + c];
    sum += v; sq += v * v;
  }
  atomicAdd(&s_acc[c], sum);          // ds_add_f32
  atomicAdd(&s_acc[C + c], sq);
  __syncthreads();
  if (threadIdx.x < 2 * C) atomicAdd(&stats[threadIdx.x], s_acc[threadIdx.x]);
}

// In-place: buf = (buf - mean) * rsqrt(var + eps) * gamma + beta + skip
template<int C>
__global__ __launch_bounds__(256) void bn_finalize(
    float* __restrict__ buf, const float* __restrict__ skip,
    const float* __restrict__ gamma, const float* __restrict__ beta,
    const float* __restrict__ stats, long N)
{
  long i = (long)blockIdx.x * blockDim.x + threadIdx.x;
  if (i >= N * C) return;
  const int c = (int)(i & (C - 1));
  const float mean = stats[c] * (1.0f / (float)N);
  const float var  = stats[C + c] * (1.0f / (float)N) - mean * mean;
  const float inv  = rsqrtf(var + BN_EPS);
  buf[i] = (buf[i] - mean) * inv * gamma[c] + beta[c] + skip[i];
}

extern "C" void kernel_launch(void* const* d_in, const int* in_sizes, int n_in,
                              void* d_out, int out_size, void* d_ws, size_t ws_size,
                              hipStream_t stream)
{
  const float* x3     = (const float*)d_in[0];
  const float* skip3  = (const float*)d_in[1];
  const float* skip2  = (const float*)d_in[2];
  const float* skip1  = (const float*)d_in[3];
  const float* W3     = (const float*)d_in[4];
  const float* W2     = (const float*)d_in[5];
  const float* W1     = (const float*)d_in[6];
  const float* gamma3 = (const float*)d_in[7];
  const float* beta3  = (const float*)d_in[8];
  const float* gamma2 = (const float*)d_in[9];
  const float* beta2  = (const float*)d_in[10];
  const float* gamma1 = (const float*)d_in[11];
  const float* beta1  = (const float*)d_in[12];
  const int* in_idx3  = (const int*)d_in[13];
  const int* out_idx3 = (const int*)d_in[14];
  const int* in_idx2  = (const int*)d_in[15];
  const int* out_idx2 = (const int*)d_in[16];
  const int* in_idx1  = (const int*)d_in[17];
  const int* out_idx1 = (const int*)d_in[18];

  const int  N3 = in_sizes[0] / 128;     // 40000
  const int  N2 = in_sizes[1] / 64;      // 160000
  const int  N1 = in_sizes[2] / 32;      // 640000
  const long N0 = (long)in_sizes[3] / 32; // 2560000

  // Workspace layout: [stats (1KB used, 4KB reserved)] [buf3 41MB] [buf2 82MB]
  float* stats3 = (float*)d_ws;          // 128 floats (sum|sumsq, C=64)
  float* stats2 = stats3 + 128;          // 64 floats
  float* stats1 = stats2 + 64;           // 64 floats
  float* buf3   = (float*)((char*)d_ws + 4096);
  float* buf2   = buf3 + (size_t)N2 * 64;
  float* out1   = (float*)d_out;

  const size_t zeroBytes = 4096 + ((size_t)N2 * 64 + (size_t)N1 * 32) * sizeof(float);
  hipMemsetAsync(d_ws, 0, zeroBytes, stream);                 // stats + buf3 + buf2
  hipMemsetAsync(d_out, 0, (size_t)out_size * sizeof(float), stream);

  // ---- stage 3: [40000,128] -> [160000,64] ----
  {
    const int tiles = KSTEN * (N3 / 16);
    up_gemm_scatter<128, 64><<<(tiles + 7) / 8, 256, 0, stream>>>(
        x3, W3, in_idx3, out_idx3, buf3, N3);
    bn_stats<64><<<2048, 256, 0, stream>>>(buf3, N2, stats3);
    const long tot = (long)N2 * 64;
    bn_finalize<64><<<(unsigned)((tot + 255) / 256), 256, 0, stream>>>(
        buf3, skip3, gamma3, beta3, stats3, (long)N2);
  }
  // ---- stage 2: [160000,64] -> [640000,32] ----
  {
    const int tiles = KSTEN * (N2 / 16);
    up_gemm_scatter<64, 32><<<(tiles + 7) / 8, 256, 0, stream>>>(
        buf3, W2, in_idx2, out_idx2, buf2, N2);
    bn_stats<32><<<2048, 256, 0, stream>>>(buf2, N1, stats2);
    const long tot = (long)N1 * 32;
    bn_finalize<32><<<(unsigned)((tot + 255) / 256), 256, 0, stream>>>(
        buf2, skip2, gamma2, beta2, stats2, (long)N1);
  }
  // ---- stage 1: [640000,32] -> [2560000,32] (accumulate directly in d_out) ----
  {
    const int tiles = KSTEN * (N1 / 16);
    up_gemm_scatter<32, 32><<<(tiles + 7) / 8, 256, 0, stream>>>(
        buf2, W1, in_idx1, out_idx1, out1, N1);
    bn_stats<32><<<4096, 256, 0, stream>>>(out1, (int)N0, stats1);
    const long tot = N0 * 32;
    bn_finalize<32><<<(unsigned)((tot + 255) / 256), 256, 0, stream>>>(
        out1, skip1, gamma1, beta1, stats1, N0);
  }
}